// CropPadding_73667279061677
// MI455X (gfx1250) — compile-verified
//
#include <hip/hip_runtime.h>

#define RES 1024
#define THRES 100

typedef __attribute__((ext_vector_type(2))) float v2f;
typedef __attribute__((ext_vector_type(8))) float v8f;

// ---------------------------------------------------------------------------
// Kernel 1: per-view foreground bbox -> per-view resample parameters (8 floats)
// params[v*8 + {0..7}] = min_h, min_w, h_n, w_n, out_h, out_w, pad_t, pad_l
// ---------------------------------------------------------------------------
__global__ __launch_bounds__(1024) void bbox_kernel(const float* __restrict__ masks,
                                                    float* __restrict__ params) {
  const int v = blockIdx.x;
  const int t = threadIdx.x;                       // 0..1023
  const float* m = masks + (size_t)v * RES * RES;

  __shared__ int s_minh, s_maxh, s_minw, s_maxw;
  if (t == 0) { s_minh = RES; s_maxh = -1; s_minw = RES; s_maxw = -1; }
  __syncthreads();

  // Column-any: thread t = column x, loop over y (fully coalesced loads).
  bool colAny = false;
  for (int y = 0; y < RES; ++y)
    colAny = colAny || (m[(size_t)y * RES + t] > 0.5f);
  if (colAny) { atomicMin(&s_minw, t); atomicMax(&s_maxw, t); }

  // Row-any: thread t = row y, float4 stream along x (L2-friendly).
  bool rowAny = false;
  const float4* row = (const float4*)(m + (size_t)t * RES);
  for (int x = 0; x < RES / 4; ++x) {
    float4 q = row[x];
    rowAny = rowAny || (q.x > 0.5f) || (q.y > 0.5f) || (q.z > 0.5f) || (q.w > 0.5f);
  }
  if (rowAny) { atomicMin(&s_minh, t); atomicMax(&s_maxh, t); }
  __syncthreads();

  if (t == 0) {
    // Empty mask -> argmax semantics of the reference give [0, RES-1].
    int minh = (s_maxh < 0) ? 0 : s_minh;
    int maxh = (s_maxh < 0) ? (RES - 1) : s_maxh;
    int minw = (s_maxw < 0) ? 0 : s_minw;
    int maxw = (s_maxw < 0) ? (RES - 1) : s_maxw;

    float mh = (float)min(max(minh - THRES, 0), RES - 1);
    float mw = (float)min(max(minw - THRES, 0), RES - 1);
    float Mh = (float)min(max(maxh + THRES, 0), RES - 1);
    float Mw = (float)min(max(maxw + THRES, 0), RES - 1);

    float hn = Mh - mh;
    float wn = Mw - mw;
    bool vert = hn >= wn;
    float oh = vert ? (float)RES : floorf(hn * (float)RES / wn);
    float ow = vert ? floorf(wn * (float)RES / hn) : (float)RES;
    float pt = floorf(((float)RES - oh) * 0.5f);
    float pl = floorf(((float)RES - ow) * 0.5f);

    float* P = params + v * 8;
    P[0] = mh; P[1] = mw; P[2] = hn; P[3] = wn;
    P[4] = oh; P[5] = ow; P[6] = pt; P[7] = pl;
  }
}

// ---------------------------------------------------------------------------
// Kernel 2: fused crop-resample (bilinear rgb + nearest mask) + rot90(k=1)
// via a padded LDS transpose tile. out layout: rgb (8,3,1024,1024) then
// mask (8,1,1024,1024).
// rot90(k=1): out[i][j] = pre[j][RES-1-i]
// ---------------------------------------------------------------------------
__global__ __launch_bounds__(1024) void resample_kernel(const float* __restrict__ rgbs,
                                                        const float* __restrict__ masks,
                                                        const float* __restrict__ params,
                                                        float* __restrict__ out) {
  const int v = blockIdx.z;
  const int ox0 = blockIdx.x * 32;   // output tile origin (rotated frame)
  const int oy0 = blockIdx.y * 32;
  const int tx = threadIdx.x;
  const int ty = threadIdx.y;

  __shared__ float tile[4][32][33];  // 33: bank-conflict-free transpose

  const float* P = params + v * 8;
  const float mh = P[0], mw = P[1], hn = P[2], wn = P[3];
  const float oh = P[4], ow = P[5], pt = P[6], pl = P[7];
  const float sxr = wn / ow;
  const float syr = hn / oh;

  // Pre-rotation coordinates covered by this tile:
  //   ry in [ox0, ox0+31] (thread ty), rx in [rx0, rx0+31] (thread tx)
  const int rx0 = (RES - 32) - oy0;
  const int ry = ox0 + ty;
  const int rx = rx0 + tx;

  const float xs = (float)rx - pl;
  const float ys = (float)ry - pt;
  const bool vx = (xs >= 0.0f) && (xs < ow);
  const bool vy = (ys >= 0.0f) && (ys < oh);
  const float valid = (vx && vy) ? 1.0f : 0.0f;

  float sx = fmaxf((xs + 0.5f) * sxr - 0.5f, 0.0f);
  float sy = fmaxf((ys + 0.5f) * syr - 0.5f, 0.0f);
  float x0 = floorf(sx);  const float fx = sx - x0;
  float y0 = floorf(sy);  const float fy = sy - y0;
  x0 = fminf(x0, wn - 1.0f);
  const float x1 = fminf(x0 + 1.0f, wn - 1.0f);
  y0 = fminf(y0, hn - 1.0f);
  const float y1 = fminf(y0 + 1.0f, hn - 1.0f);

  const int gx0 = (int)(x0 + mw);
  const int gx1 = (int)(x1 + mw);
  const int gy0 = (int)(y0 + mh);
  const int gy1 = (int)(y1 + mh);

  const size_t plane = (size_t)RES * RES;
  const float* img = rgbs + (size_t)v * 3 * plane;
#pragma unroll
  for (int c = 0; c < 3; ++c) {
    const float* ch = img + (size_t)c * plane;
    const float a = ch[(size_t)gy0 * RES + gx0];
    const float b = ch[(size_t)gy0 * RES + gx1];
    const float d = ch[(size_t)gy1 * RES + gx0];
    const float e = ch[(size_t)gy1 * RES + gx1];
    const float top = a * (1.0f - fx) + b * fx;
    const float bot = d * (1.0f - fx) + e * fx;
    tile[c][ty][tx] = (top * (1.0f - fy) + bot * fy) * valid;
  }

  // mask: nearest
  const float nxf = fminf(fmaxf(floorf(xs * sxr), 0.0f), wn - 1.0f) + mw;
  const float nyf = fminf(fmaxf(floorf(ys * syr), 0.0f), hn - 1.0f) + mh;
  const int nx = (int)nxf;
  const int ny = (int)nyf;
  tile[3][ty][tx] = masks[(size_t)v * plane + (size_t)ny * RES + nx] * valid;

  __syncthreads();

  // Rotated, coalesced stores: out[oy0+ty][ox0+tx] = tile[tx][31-ty]
  const size_t opix = (size_t)(oy0 + ty) * RES + (size_t)(ox0 + tx);
#pragma unroll
  for (int c = 0; c < 3; ++c)
    out[((size_t)v * 3 + c) * plane + opix] = tile[c][tx][31 - ty];
  float* mask_out = out + (size_t)8 * 3 * plane;
  mask_out[(size_t)v * plane + opix] = tile[3][tx][31 - ty];
}

// ---------------------------------------------------------------------------
// Kernel 3: kc_new = R @ (A @ kc) for all 8 views.
// X = A@kc computed scalarly (A is affine-diagonal), then the shared-R
// product R(3x3) @ X(3x24) is one 16x4 * 4x16 f32 GEMM -> two
// V_WMMA_F32_16X16X4_F32 calls over the 24 stacked columns.
// Single wave32, full EXEC as WMMA requires.
// ---------------------------------------------------------------------------
__global__ __launch_bounds__(32) void kc_wmma_kernel(const float* __restrict__ kc,
                                                     const float* __restrict__ params,
                                                     float* __restrict__ out_kc) {
  const int lane = threadIdx.x;  // 0..31

  // A operand = R padded to 16x4. R = [[0,1,0],[-1,0,1024],[0,0,1]] (ROT_DEG=1).
  // Layout (ISA 7.12.2, 32-bit A 16x4): lanes 0-15 hold K=0 (v0), K=1 (v1) for M=lane;
  // lanes 16-31 hold K=2 (v0), K=3 (v1) for M=lane-16.
  float a0 = 0.0f, a1 = 0.0f;
  if (lane < 16) {
    if (lane == 0)      { a0 = 0.0f;  a1 = 1.0f; }
    else if (lane == 1) { a0 = -1.0f; a1 = 0.0f; }
    // lane==2: R[2][0]=0, R[2][1]=0
  } else {
    const int M = lane - 16;
    if (M == 1) a0 = 1024.0f;   // R[1][2]
    else if (M == 2) a0 = 1.0f; // R[2][2]
    // K=3 column is zero padding
  }
  const v2f A = {a0, a1};

  // X[i][n], n = v*3 + j, i row of A@kc:
  //   X0 = sxk*kc_row0 + tx*kc_row2 ; X1 = syk*kc_row1 + ty*kc_row2 ; X2 = kc_row2
  auto xval = [&](int i, int n) -> float {
    if (n >= 24) return 0.0f;
    const int vv = n / 3, j = n % 3;
    const float* P = params + vv * 8;
    const float mh = P[0], mw = P[1], hn = P[2], wn = P[3];
    const float oh = P[4], ow = P[5], pt = P[6], pl = P[7];
    const float sxk = ow / wn, syk = oh / hn;
    const float txv = pl - mw * sxk, tyv = pt - mh * syk;
    const float k0 = kc[vv * 9 + 0 + j];
    const float k1 = kc[vv * 9 + 3 + j];
    const float k2 = kc[vv * 9 + 6 + j];
    if (i == 0) return sxk * k0 + txv * k2;
    if (i == 1) return syk * k1 + tyv * k2;
    return k2;
  };

#pragma unroll
  for (int off = 0; off < 32; off += 16) {
    // B operand 4x16 (32-bit): v0: lanes 0-15 -> K=0 N=lane, lanes 16-31 -> K=2 N=lane-16;
    //                          v1: lanes 0-15 -> K=1,        lanes 16-31 -> K=3 (zero pad).
    float b0, b1;
    if (lane < 16) {
      b0 = xval(0, lane + off);
      b1 = xval(1, lane + off);
    } else {
      b0 = xval(2, lane - 16 + off);
      b1 = 0.0f;
    }
    const v2f B = {b0, b1};
    v8f C = {};
    C = __builtin_amdgcn_wmma_f32_16x16x4_f32(
        /*neg_a=*/false, A, /*neg_b=*/false, B,
        /*c_mod=*/(short)0, C, /*reuse_a=*/false, /*reuse_b=*/false);
    // D 16x16: VGPR g, lanes 0-15 -> M=g, N=lane. Rows 0..2 are kc_new rows.
    if (lane < 16) {
      const int n = lane + off;
      if (n < 24) {
        const int vv = n / 3, j = n % 3;
        out_kc[vv * 9 + 0 + j] = C[0];
        out_kc[vv * 9 + 3 + j] = C[1];
        out_kc[vv * 9 + 6 + j] = C[2];
      }
    }
  }
}

// ---------------------------------------------------------------------------
extern "C" void kernel_launch(void* const* d_in, const int* in_sizes, int n_in,
                              void* d_out, int out_size, void* d_ws, size_t ws_size,
                              hipStream_t stream) {
  const float* rgbs  = (const float*)d_in[0];   // (8,3,1024,1024)
  const float* masks = (const float*)d_in[1];   // (8,1,1024,1024)
  const float* kc    = (const float*)d_in[2];   // (8,3,3)
  float* out = (float*)d_out;                   // rgb | mask | kc_new
  float* params = (float*)d_ws;                 // 8 views * 8 floats

  bbox_kernel<<<8, 1024, 0, stream>>>(masks, params);

  dim3 grid(RES / 32, RES / 32, 8);
  dim3 block(32, 32, 1);
  resample_kernel<<<grid, block, 0, stream>>>(rgbs, masks, params, out);

  const size_t kc_base = (size_t)8 * 4 * RES * RES;  // 8*(3+1) planes
  kc_wmma_kernel<<<1, 32, 0, stream>>>(kc, params, out + kc_base);
}